// MojoSimpleSampler_65970697667083
// MI455X (gfx1250) — compile-verified
//
#include <hip/hip_runtime.h>
#include <hip/hip_bf16.h>
#include <float.h>
#include <stdint.h>

#define TOPP    0.9f
#define THREADS 256
#define WAVES   (THREADS / 32)
#define NB      1024                 // histogram bins per zoom level
#define CHUNK   (NB / THREADS)       // 4 bins per thread
#define LEVELS  3                    // 88 / 1024^3 ~ 8e-8 < f32 ulp at boundary
#define CAP     64                   // survivor collect capacity
#define RANGE   88.0f                // exp underflows below rmax-88

typedef float v2f __attribute__((ext_vector_type(2)));
typedef float v8f __attribute__((ext_vector_type(8)));

// ---------------- CDNA5 async load-to-LDS (ASYNCcnt-tracked DMA path) -------
__device__ __forceinline__ void async_ld_b128(uint32_t lds_byte, const void* gptr) {
#if defined(__AMDGCN__)
  asm volatile("global_load_async_to_lds_b128 %0, %1, off"
               :: "v"(lds_byte), "v"(gptr) : "memory");
#endif
}
__device__ __forceinline__ void wait_async_le1() {
#if defined(__AMDGCN__)
  asm volatile("s_wait_asynccnt 0x1" ::: "memory");
#endif
}
__device__ __forceinline__ void wait_async_le0() {
#if defined(__AMDGCN__)
  asm volatile("s_wait_asynccnt 0x0" ::: "memory");
#endif
}

// ---------------- block-wide f32 sum, reduced on the matrix unit ------------
// Lane L<16 contributes A[L][0], lane L>=16 contributes A[L-16][2] (16x4 f32 A
// layout); B = ones(4x16) => D[m][n] = sum_k A[m][k]. Lanes 0-15 end up with
// sum(rows 0-7) in d[0..7], lanes 16-31 with sum(rows 8-15).
__device__ __forceinline__ float block_sum(float v, float* wsum) {
  float s;
#if __has_builtin(__builtin_amdgcn_wmma_f32_16x16x4_f32)
  v2f a; a.x = v;    a.y = 0.0f;
  v2f b; b.x = 1.0f; b.y = 1.0f;
  v8f c = {};
  c = __builtin_amdgcn_wmma_f32_16x16x4_f32(
      false, a, false, b, (short)0, c, false, false);
  s = ((c[0] + c[1]) + (c[2] + c[3])) + ((c[4] + c[5]) + (c[6] + c[7]));
  s += __shfl_xor(s, 16, 32);
#else
  s = v;
  for (int m = 16; m >= 1; m >>= 1) s += __shfl_xor(s, m, 32);
#endif
  const int lane = threadIdx.x & 31;
  const int wv   = threadIdx.x >> 5;
  if (lane == 0) wsum[wv] = s;
  __syncthreads();
  float tot = 0.0f;
#pragma unroll
  for (int w = 0; w < WAVES; ++w) tot += wsum[w];
  __syncthreads();
  return tot;
}

// Monotone clamp-map: defines bin membership exactly (no edge-rounding drift).
__device__ __forceinline__ int bin_of(float x, float glo, float scl) {
  int b = (int)((x - glo) * scl);
  b = b < 0 ? 0 : b;
  b = b > (NB - 1) ? (NB - 1) : b;
  return b;
}

__device__ __forceinline__ uint64_t splitmix64(uint64_t z) {
  z += 0x9E3779B97F4A7C15ULL;
  z = (z ^ (z >> 30)) * 0xBF58476D1CE4E5B9ULL;
  z = (z ^ (z >> 27)) * 0x94D049BB133111EBULL;
  return z ^ (z >> 31);
}

// Histogram-zoom select: finds the element where the descending cumulative
// exp-mass crosses `target_abs + target_frac*Z` (Z = total mass, measured via
// the WMMA block reduce at level 0). Returns its flat index; *kept_mass gets
// the cumulative mass through (and including) that element.
__device__ int zoom_select(const float4* __restrict__ x4, int n4, float rmax,
                           float target_abs, float target_frac, int fallback_idx,
                           float* bins, float* scan, float* wsum,
                           float* cf, int* ci, int* sh_i, float* sh_f,
                           float* kept_mass) {
  const int tid = threadIdx.x;
  float glo[LEVELS], scl[LEVELS];
  int   cho[LEVELS];
  glo[0] = rmax - RANGE;
  scl[0] = (float)NB / RANGE;      // deepest scale ~1.2e7: (x-glo)*scl < 2^31
  float A = 0.0f;                  // exact mass strictly above current node
  float target = target_abs;

#pragma unroll
  for (int L = 0; L < LEVELS; ++L) {
    for (int b = tid; b < NB; b += THREADS) bins[b] = 0.0f;
    if (tid == 0) sh_i[0] = -1;
    __syncthreads();

    float local = 0.0f;
    for (int i = tid; i < n4; i += THREADS) {
      float4 v = x4[i];
      float xs[4] = {v.x, v.y, v.z, v.w};
#pragma unroll
      for (int c = 0; c < 4; ++c) {
        float xx = xs[c];
        if (L == 0) {
          float w = __expf(xx - rmax);
          local += w;
          atomicAdd(&bins[bin_of(xx, glo[0], scl[0])], w);   // ds_add_f32
        } else {
          bool ok = true;
#pragma unroll
          for (int j = 0; j < LEVELS; ++j)
            if (j < L) ok = ok && (bin_of(xx, glo[j], scl[j]) == cho[j]);
          if (ok)
            atomicAdd(&bins[bin_of(xx, glo[L], scl[L])], __expf(xx - rmax));
        }
      }
    }
    if (L == 0) {
      float Z = block_sum(local, wsum);          // WMMA reduce (+barrier)
      target = target_abs + target_frac * Z;
    }
    __syncthreads();                             // histogram complete

    // Suffix scan over 4-bin chunk sums (Hillis-Steele, high-to-low).
    float csum = bins[CHUNK * tid] + bins[CHUNK * tid + 1] +
                 bins[CHUNK * tid + 2] + bins[CHUNK * tid + 3];
    scan[tid] = csum;
    __syncthreads();
    for (int off = 1; off < THREADS; off <<= 1) {
      float add = (tid + off < THREADS) ? scan[tid + off] : 0.0f;
      __syncthreads();
      scan[tid] += add;
      __syncthreads();
    }

    // Unique crossing bin b: A + excl(b) <= target < A + excl(b) + bins[b].
    float r = (tid + 1 < THREADS) ? scan[tid + 1] : 0.0f;
#pragma unroll
    for (int k = CHUNK - 1; k >= 0; --k) {
      int b = CHUNK * tid + k;
      float bm = bins[b];
      if (bm > 0.0f && (A + r) <= target && target < (A + r + bm)) {
        sh_i[0] = b; sh_f[0] = A + r;
      }
      r += bm;
    }
    __syncthreads();
    int b = sh_i[0];
    if (b >= 0) A = sh_f[0]; else b = 0;         // ulp-degenerate fallback
    cho[L] = b;
    if (L + 1 < LEVELS) {
      glo[L + 1] = glo[L] + (float)b / scl[L];
      scl[L + 1] = scl[L] * (float)NB;
    }
    __syncthreads();
  }

  // Collect survivors of all LEVELS bin tests (usually exactly 1 token).
  if (tid == 0) sh_i[1] = 0;
  __syncthreads();
  for (int i = tid; i < n4; i += THREADS) {
    float4 v = x4[i];
    float xs[4] = {v.x, v.y, v.z, v.w};
#pragma unroll
    for (int c = 0; c < 4; ++c) {
      float xx = xs[c];
      bool ok = true;
#pragma unroll
      for (int j = 0; j < LEVELS; ++j)
        ok = ok && (bin_of(xx, glo[j], scl[j]) == cho[j]);
      if (ok) {
        int k = atomicAdd(&sh_i[1], 1);          // ds_add_rtn_u32
        if (k < CAP) { cf[k] = xx; ci[k] = i * 4 + c; }
      }
    }
  }
  __syncthreads();

  if (tid == 0) {
    int n = sh_i[1]; n = n > CAP ? CAP : n;
    // Stable order matching argsort(-p): descending value, ties by asc index.
    for (int a = 1; a < n; ++a) {
      float fv = cf[a]; int fi = ci[a]; int p = a - 1;
      while (p >= 0 && (cf[p] < fv || (cf[p] == fv && ci[p] > fi))) {
        cf[p + 1] = cf[p]; ci[p + 1] = ci[p]; --p;
      }
      cf[p + 1] = fv; ci[p + 1] = fi;
    }
    float Ac = A; int sel = fallback_idx;
    float km = fmaxf(A, 1e-30f); bool found = false;
    for (int j = 0; j < n; ++j) {
      float w = __expf(cf[j] - rmax);
      if (!found && Ac <= target && target < Ac + w) {
        sel = ci[j]; km = Ac + w; found = true;
      }
      Ac += w;
    }
    if (!found && n > 0) { sel = ci[n - 1]; km = Ac; }
    sh_i[2] = sel; sh_f[1] = km;
  }
  __syncthreads();
  *kept_mass = sh_f[1];
  int res = sh_i[2];
  __syncthreads();
  return res;
}

__global__ __launch_bounds__(THREADS)
void topp_sample_kernel(const float* __restrict__ logits,
                        int* __restrict__ out, int V, uint64_t seed) {
  const int row = blockIdx.x;
  const int tid = threadIdx.x;
  const float*  x  = logits + (size_t)row * (size_t)V;
  const float4* x4 = (const float4*)x;
  const int n4 = V >> 2;                         // V % 4 == 0 (128000)

  __shared__ float4   stage[2][THREADS];         // async DMA double buffer 8KB
  __shared__ float    bins[NB];                  // zoom histogram 4KB
  __shared__ float    scan[THREADS];
  __shared__ float    red_f[THREADS];
  __shared__ int      red_i[THREADS];
  __shared__ float    wsum[WAVES];
  __shared__ float    cf[CAP];
  __shared__ int      ci[CAP];
  __shared__ int      sh_i[4];
  __shared__ float    sh_f[2];

  // ---- Phase 1: row max + argmax. The only HBM-cold pass: stage tiles into
  // LDS with async load-to-LDS, double-buffered on ASYNCcnt.
  float lmax = -FLT_MAX; int lidx = 0;
  const int ntiles = n4 / THREADS;               // 125
  const uint32_t slot0 = (uint32_t)(uintptr_t)&stage[0][tid];
  const uint32_t slot1 = (uint32_t)(uintptr_t)&stage[1][tid];
  if (ntiles > 0) async_ld_b128(slot0, (const void*)(x4 + tid));
  for (int k = 0; k < ntiles; ++k) {
    if (k + 1 < ntiles) {
      async_ld_b128((k & 1) ? slot0 : slot1,
                    (const void*)(x4 + (size_t)(k + 1) * THREADS + tid));
      wait_async_le1();
    } else {
      wait_async_le0();
    }
    float4 v = stage[k & 1][tid];                // own slot: no barrier needed
    int base = (k * THREADS + tid) * 4;
    if (v.x > lmax) { lmax = v.x; lidx = base;     }
    if (v.y > lmax) { lmax = v.y; lidx = base + 1; }
    if (v.z > lmax) { lmax = v.z; lidx = base + 2; }
    if (v.w > lmax) { lmax = v.w; lidx = base + 3; }
  }
  for (int i = ntiles * THREADS + tid; i < n4; i += THREADS) {  // generic tail
    float4 v = x4[i];
    int base = i * 4;
    if (v.x > lmax) { lmax = v.x; lidx = base;     }
    if (v.y > lmax) { lmax = v.y; lidx = base + 1; }
    if (v.z > lmax) { lmax = v.z; lidx = base + 2; }
    if (v.w > lmax) { lmax = v.w; lidx = base + 3; }
  }
  red_f[tid] = lmax; red_i[tid] = lidx;
  __syncthreads();
  for (int s = THREADS / 2; s > 0; s >>= 1) {
    if (tid < s) {
      float fo = red_f[tid + s]; int io = red_i[tid + s];
      if (fo > red_f[tid] || (fo == red_f[tid] && io < red_i[tid])) {
        red_f[tid] = fo; red_i[tid] = io;
      }
    }
    __syncthreads();
  }
  const float rmax     = red_f[0];
  const int   argmax_i = red_i[0];
  __syncthreads();

  // ---- Phase 2: nucleus boundary token via histogram zoom at p*Z.
  // K = cumulative mass through the boundary token = renormalization mass.
  float K;
  (void)zoom_select(x4, n4, rmax, 0.0f, TOPP, argmax_i,
                    bins, scan, wsum, cf, ci, sh_i, sh_f, &K);

  // ---- Phase 3: categorical draw = inverse-CDF at u*K over the same order.
  // (Crossing element has cum-above < K, so it is in the kept set.)
  uint64_t h = splitmix64(seed ^ (0x9E3779B97F4A7C15ULL * (uint64_t)(row + 1)));
  float u = (float)((double)(h >> 11) * (1.0 / 9007199254740992.0));
  u = fminf(fmaxf(u, 1e-7f), 1.0f - 1e-7f);
  float dummy;
  int tok = zoom_select(x4, n4, rmax, u * K, 0.0f, argmax_i,
                        bins, scan, wsum, cf, ci, sh_i, sh_f, &dummy);

  if (tid == 0) out[row] = tok;
}

extern "C" void kernel_launch(void* const* d_in, const int* in_sizes, int n_in,
                              void* d_out, int out_size, void* d_ws, size_t ws_size,
                              hipStream_t stream) {
  (void)n_in; (void)d_ws; (void)ws_size;
  const float* logits = (const float*)d_in[0];
  int* out = (int*)d_out;                        // one int32 token id per row
  const int B = out_size;                        // 128
  const int V = in_sizes[0] / B;                 // 128000
  topp_sample_kernel<<<dim3(B), dim3(THREADS), 0, stream>>>(logits, out, V, 42ULL);
}